// TransitionUp_56624848831037
// MI455X (gfx1250) — compile-verified
//
#include <hip/hip_runtime.h>
#include <hip/hip_bf16.h>

typedef __attribute__((ext_vector_type(2))) float v2f;
typedef __attribute__((ext_vector_type(8))) float v8f;

#define NCH 256          // Cout for both branches
#define ROWS 32768       // B*N2
#define NCOARSE 2048     // N1 per batch
#define NFINE 8192       // N2 per batch
#define NBATCH 4
#define STATS_BLOCKS 64
#define KCHUNK 64        // K-chunk staged in LDS per iteration

// ---------------------------------------------------------------------------
// Kernel 1: 3-nearest-neighbor + inverse-distance weights.
// One block = 256 fine points of one batch; coarse xyz staged in LDS (24KB).
// ---------------------------------------------------------------------------
__global__ void __launch_bounds__(256) knn3_kernel(
    const float* __restrict__ xyz2, const float* __restrict__ xyz1,
    int* __restrict__ idx, float* __restrict__ wgt) {
  __shared__ float sx[NCOARSE], sy[NCOARSE], sz[NCOARSE];
  const int b = blockIdx.x >> 5;      // 32 blocks per batch
  const int tile = blockIdx.x & 31;
  const float* cp = xyz1 + (size_t)b * NCOARSE * 3;
  for (int j = threadIdx.x; j < NCOARSE; j += 256) {
    sx[j] = cp[j * 3 + 0];
    sy[j] = cp[j * 3 + 1];
    sz[j] = cp[j * 3 + 2];
  }
  __syncthreads();

  const int n = tile * 256 + threadIdx.x;      // fine point index in batch
  const size_t g = (size_t)b * NFINE + n;      // flat row
  const float px = xyz2[g * 3 + 0];
  const float py = xyz2[g * 3 + 1];
  const float pz = xyz2[g * 3 + 2];

  float d0 = 3.0e38f, d1 = 3.0e38f, d2 = 3.0e38f;
  int i0 = 0, i1 = 0, i2 = 0;
  for (int j = 0; j < NCOARSE; ++j) {
    const float dx = px - sx[j], dy = py - sy[j], dz = pz - sz[j];
    const float d = dx * dx + dy * dy + dz * dz;
    if (d < d2) {
      if (d < d1) {
        if (d < d0) { d2 = d1; i2 = i1; d1 = d0; i1 = i0; d0 = d; i0 = j; }
        else        { d2 = d1; i2 = i1; d1 = d;  i1 = j; }
      } else        { d2 = d;  i2 = j; }
    }
  }
  const float r0 = 1.0f / (d0 + 1e-8f);
  const float r1 = 1.0f / (d1 + 1e-8f);
  const float r2 = 1.0f / (d2 + 1e-8f);
  const float inv = 1.0f / (r0 + r1 + r2);
  idx[g * 3 + 0] = b * NCOARSE + i0;   // flat coarse-row index
  idx[g * 3 + 1] = b * NCOARSE + i1;
  idx[g * 3 + 2] = b * NCOARSE + i2;
  wgt[g * 3 + 0] = r0 * inv;
  wgt[g * 3 + 1] = r1 * inv;
  wgt[g * 3 + 2] = r2 * inv;
}

// ---------------------------------------------------------------------------
// Kernel 2: fp32 WMMA GEMM  C[M,256] = A[M,K] @ B[K,256] + bias.
// Block tile: 256 rows x 64 cols, 8 waves stacked in M.
// B is staged in LDS one 64x64 K-chunk at a time, in a (k-pair, n) swizzled
// layout: element (kk, n) -> sB[(kk>>1)*128 + n*2 + (kk&1)], so the fragment
// pair {B[ka][n], B[ka+1][n]} (ka even) is one aligned ds_load_b64 and lanes
// 0..15 hit 16 consecutive 8-byte slots (conflict-free banking).
// Per wave: 32x64 output tile, 8 v_wmma_f32_16x16x4_f32 per K=4 step.
// Fragment layouts per CDNA5 ISA 7.12.2 (f32 16x16x4 shape):
//   A/B frag (2 VGPRs): element j of lane l holds K = j + 2*(l>>4)
//   C/D (8 VGPRs):      element r of lane l holds M = r + 8*(l>>4), N = l&15
// ---------------------------------------------------------------------------
__global__ void __launch_bounds__(256) gemm_wmma_f32(
    const float* __restrict__ A, const float* __restrict__ B,
    const float* __restrict__ bias, float* __restrict__ C,
    int M, int K) {
  const int N = NCH;
  __shared__ float sB[KCHUNK * 64];    // 16 KB, swizzled

  const int tid  = threadIdx.x;
  const int wid  = tid >> 5;
  const int lane = tid & 31;
  const int half = lane >> 4;
  const int c16  = lane & 15;

  const int nTiles = N / 64;                   // 4
  const int n0 = (blockIdx.x % nTiles) * 64;
  const int m0 = (blockIdx.x / nTiles) * 256 + wid * 32;

  v8f acc[2][4] = {};
  const float* Arow0 = A + (size_t)(m0 + c16) * K;
  const float* Arow1 = A + (size_t)(m0 + 16 + c16) * K;

  for (int kc = 0; kc < K; kc += KCHUNK) {
    // Cooperative load of B[kc..kc+63, n0..n0+63] into swizzled LDS.
    // 1024 float4 elements, 4 per thread, fully coalesced global_load_b128.
#pragma unroll
    for (int i = 0; i < 4; ++i) {
      const int e  = tid + i * 256;            // 0..1023
      const int kk = e >> 4;                   // 0..63
      const int nn = (e & 15) * 4;             // 0..60
      const float4 v = *(const float4*)(B + (size_t)(kc + kk) * N + n0 + nn);
      float* s = sB + (kk >> 1) * 128 + (kk & 1);
      s[(nn + 0) * 2] = v.x;
      s[(nn + 1) * 2] = v.y;
      s[(nn + 2) * 2] = v.z;
      s[(nn + 3) * 2] = v.w;
    }
    __syncthreads();

    if (kc + KCHUNK < K) {                     // prefetch next A chunk
      __builtin_prefetch(Arow0 + kc + KCHUNK, 0, 1);
      __builtin_prefetch(Arow1 + kc + KCHUNK, 0, 1);
    }

#pragma unroll 2
    for (int k = 0; k < KCHUNK; k += 4) {
      const int ka = k + 2 * half;             // even
      const v2f a0 = *(const v2f*)(Arow0 + kc + ka);
      const v2f a1 = *(const v2f*)(Arow1 + kc + ka);
      const float* sBk = sB + (ka >> 1) * 128 + c16 * 2;
      v2f bf[4];
#pragma unroll
      for (int j = 0; j < 4; ++j)
        bf[j] = *(const v2f*)(sBk + 32 * j);   // {B[ka][n], B[ka+1][n]}
#pragma unroll
      for (int j = 0; j < 4; ++j) {
        acc[0][j] = __builtin_amdgcn_wmma_f32_16x16x4_f32(
            false, a0, false, bf[j], (short)0, acc[0][j], false, false);
        acc[1][j] = __builtin_amdgcn_wmma_f32_16x16x4_f32(
            false, a1, false, bf[j], (short)0, acc[1][j], false, false);
      }
    }
    __syncthreads();
  }

#pragma unroll
  for (int i = 0; i < 2; ++i) {
#pragma unroll
    for (int j = 0; j < 4; ++j) {
      const int col = n0 + 16 * j + c16;
      const float bv = bias[col];
      float* Cp = C + (size_t)(m0 + 16 * i + 8 * half) * N + col;
#pragma unroll
      for (int r = 0; r < 8; ++r)
        Cp[(size_t)r * N] = acc[i][j][r] + bv;
    }
  }
}

// ---------------------------------------------------------------------------
// Kernel 3: weighted 3-NN gather of projected coarse features.
// y2[row,c] = sum_k w_k * proj[idx_k, c]   (bias b2 already folded into proj,
// valid because the weights are normalized to sum to 1).
// ---------------------------------------------------------------------------
__global__ void __launch_bounds__(256) combine_kernel(
    const float* __restrict__ proj, const int* __restrict__ idx,
    const float* __restrict__ wgt, float* __restrict__ y2) {
  const int row = blockIdx.x;
  const int c = threadIdx.x;
  const int i0 = idx[(size_t)row * 3 + 0];
  const int i1 = idx[(size_t)row * 3 + 1];
  const int i2 = idx[(size_t)row * 3 + 2];
  const float w0 = wgt[(size_t)row * 3 + 0];
  const float w1 = wgt[(size_t)row * 3 + 1];
  const float w2 = wgt[(size_t)row * 3 + 2];
  y2[(size_t)row * NCH + c] =
      w0 * proj[(size_t)i0 * NCH + c] +
      w1 * proj[(size_t)i1 * NCH + c] +
      w2 * proj[(size_t)i2 * NCH + c];
}

// ---------------------------------------------------------------------------
// Kernel 4: deterministic per-channel partial sums / sums-of-squares.
// partial layout: [block][4][256] = (sum1, sq1, sum2, sq2)
// ---------------------------------------------------------------------------
__global__ void __launch_bounds__(256) stats_kernel(
    const float* __restrict__ y1, const float* __restrict__ y2,
    float* __restrict__ partial) {
  const int c = threadIdx.x;
  const int rowsPerBlock = ROWS / STATS_BLOCKS;  // 512
  const size_t r0 = (size_t)blockIdx.x * rowsPerBlock;
  float s1 = 0.f, q1 = 0.f, s2 = 0.f, q2 = 0.f;
  for (int r = 0; r < rowsPerBlock; ++r) {
    const float a = y1[(r0 + r) * NCH + c];
    const float b = y2[(r0 + r) * NCH + c];
    s1 += a; q1 += a * a;
    s2 += b; q2 += b * b;
  }
  float* p = partial + (size_t)blockIdx.x * (4 * NCH);
  p[c]           = s1;
  p[NCH + c]     = q1;
  p[2 * NCH + c] = s2;
  p[3 * NCH + c] = q2;
}

// ---------------------------------------------------------------------------
// Kernel 5: reduce partials, build per-channel BN scale/shift.
// stats out: [scale1, shift1, scale2, shift2] x 256
// ---------------------------------------------------------------------------
__global__ void __launch_bounds__(256) prep_kernel(
    const float* __restrict__ partial, const float* __restrict__ g1,
    const float* __restrict__ be1, const float* __restrict__ g2,
    const float* __restrict__ be2, float* __restrict__ stats) {
  const int c = threadIdx.x;
  float s1 = 0.f, q1 = 0.f, s2 = 0.f, q2 = 0.f;
  for (int k = 0; k < STATS_BLOCKS; ++k) {
    const float* p = partial + (size_t)k * (4 * NCH);
    s1 += p[c];
    q1 += p[NCH + c];
    s2 += p[2 * NCH + c];
    q2 += p[3 * NCH + c];
  }
  const float invN = 1.0f / (float)ROWS;
  const float mu1 = s1 * invN;
  const float var1 = q1 * invN - mu1 * mu1;
  const float mu2 = s2 * invN;
  const float var2 = q2 * invN - mu2 * mu2;
  const float sc1 = g1[c] * rsqrtf(var1 + 1e-5f);
  const float sc2 = g2[c] * rsqrtf(var2 + 1e-5f);
  stats[c]           = sc1;
  stats[NCH + c]     = be1[c] - mu1 * sc1;
  stats[2 * NCH + c] = sc2;
  stats[3 * NCH + c] = be2[c] - mu2 * sc2;
}

// ---------------------------------------------------------------------------
// Kernel 6: fused BN + ReLU + add.
// ---------------------------------------------------------------------------
__global__ void __launch_bounds__(256) finalize_kernel(
    const float* __restrict__ y1, const float* __restrict__ y2,
    const float* __restrict__ stats, float* __restrict__ out) {
  const size_t i = (size_t)blockIdx.x * 256 + threadIdx.x;
  const int c = (int)(i & (NCH - 1));
  const float a = fmaxf(fmaf(y1[i], stats[c],           stats[NCH + c]),     0.0f);
  const float b = fmaxf(fmaf(y2[i], stats[2 * NCH + c], stats[3 * NCH + c]), 0.0f);
  out[i] = a + b;
}

// ---------------------------------------------------------------------------
extern "C" void kernel_launch(void* const* d_in, const int* in_sizes, int n_in,
                              void* d_out, int out_size, void* d_ws, size_t ws_size,
                              hipStream_t stream) {
  const float* px1  = (const float*)d_in[0];   // [4,2048,512]
  const float* xyz1 = (const float*)d_in[1];   // [4,2048,3]
  const float* px2  = (const float*)d_in[2];   // [4,8192,256]
  const float* xyz2 = (const float*)d_in[3];   // [4,8192,3]
  const float* W1   = (const float*)d_in[4];   // [256,256]
  const float* b1   = (const float*)d_in[5];
  const float* g1   = (const float*)d_in[6];
  const float* be1  = (const float*)d_in[7];
  const float* W2   = (const float*)d_in[8];   // [512,256]
  const float* b2   = (const float*)d_in[9];
  const float* g2   = (const float*)d_in[10];
  const float* be2  = (const float*)d_in[11];
  float* out = (float*)d_out;

  // Workspace layout (floats)
  float* F       = (float*)d_ws;
  float* stats   = F;                                   // 4*256 scale/shift
  float* partial = stats + 4 * NCH;                     // 64*4*256
  float* wgt     = partial + STATS_BLOCKS * 4 * NCH;    // 32768*3
  float* proj    = wgt + (size_t)ROWS * 3;              // 8192*256
  float* y1      = proj + (size_t)NBATCH * NCOARSE * NCH;  // 32768*256
  float* y2      = y1 + (size_t)ROWS * NCH;             // 32768*256
  int*   idx     = (int*)(y2 + (size_t)ROWS * NCH);     // 32768*3

  // 1. 3-NN + weights
  knn3_kernel<<<NBATCH * (NFINE / 256), 256, 0, stream>>>(xyz2, xyz1, idx, wgt);

  // 2. proj = px1 @ W2 + b2   (coarse rows only: 4x cheaper than interp GEMM)
  {
    const int M = NBATCH * NCOARSE;   // 8192, K = 512
    gemm_wmma_f32<<<M / 64, 256, 0, stream>>>(px1, W2, b2, proj, M, 512);
  }
  // 3. y1 = px2 @ W1 + b1
  {
    const int M = ROWS;               // 32768, K = 256
    gemm_wmma_f32<<<M / 64, 256, 0, stream>>>(px2, W1, b1, y1, M, 256);
  }
  // 4. y2 = weighted gather of proj
  combine_kernel<<<ROWS, 256, 0, stream>>>(proj, idx, wgt, y2);

  // 5. BN statistics (deterministic two-level reduction)
  stats_kernel<<<STATS_BLOCKS, 256, 0, stream>>>(y1, y2, partial);
  prep_kernel<<<1, 256, 0, stream>>>(partial, g1, be1, g2, be2, stats);

  // 6. out = relu(bn1(y1)) + relu(bn2(y2))
  finalize_kernel<<<out_size / 256, 256, 0, stream>>>(y1, y2, stats, out);
}